// PointsToObjects_57707180589074
// MI455X (gfx1250) — compile-verified
//
#include <hip/hip_runtime.h>

typedef unsigned int u32;
typedef unsigned long long u64;

#define B        16
#define C_TOT    84
#define C_CLS    80
#define HW       65536                 // 256*256
#define NPB      (C_CLS * HW)          // 5,242,880 class elems per batch
#define HBS      (C_TOT * HW)          // heat batch stride in floats
#define TOPK     100
#define MIN_CONF 0.1f

#define NBINS    8192                  // top 13 bits of ordered-uint
#define BIN_SH   19
#define ELEMS    65536                 // floats handled per block in scan passes
#define BPB      (NPB / ELEMS)         // 80 blocks per batch
#define NT       256
#define CAP      8192                  // candidate capacity per batch (64KB LDS sort)
#define KEEP     128

// gfx1250 async global->LDS copies (ASYNCcnt path, CDNA5 ISA)
// Signatures (from clang diagnostics):
//   b128: (v4i addrspace(1)*, v4i addrspace(3)*, imm int, imm int)
//   b64 : (v2i addrspace(1)*, v2i addrspace(3)*, imm int, imm int)
#if __has_builtin(__builtin_amdgcn_global_load_async_to_lds_b128) && \
    __has_builtin(__builtin_amdgcn_global_load_async_to_lds_b64)  && \
    __has_builtin(__builtin_amdgcn_s_wait_asynccnt)
#define HAS_ASYNC_LDS 1
typedef int v4i __attribute__((ext_vector_type(4)));
typedef int v2i __attribute__((ext_vector_type(2)));
typedef __attribute__((address_space(1))) v4i gv4i;  // global
typedef __attribute__((address_space(3))) v4i lv4i;  // LDS
typedef __attribute__((address_space(1))) v2i gv2i;
typedef __attribute__((address_space(3))) v2i lv2i;
#else
#define HAS_ASYNC_LDS 0
#endif

// Order-preserving map: float bits -> uint such that uint order == float order.
__device__ __forceinline__ u32 order_map(u32 bits) {
  return (bits & 0x80000000u) ? ~bits : (bits | 0x80000000u);
}

// ---------------------------------------------------------------------------
// Pass 1: per-batch histogram of order_map(value) >> BIN_SH
// grid = (BPB, B), block = NT.  Streaming b128 loads -> VALU -> LDS atomics.
// ---------------------------------------------------------------------------
__global__ __launch_bounds__(NT)
void hist_kernel(const float4* __restrict__ heat4, u32* __restrict__ ghist) {
  __shared__ u32 lh[NBINS];
  const int tid = threadIdx.x;
  const int batch = blockIdx.y;
  for (int i = tid; i < NBINS; i += NT) lh[i] = 0;
  __syncthreads();

  const float4* src = heat4 + (size_t)batch * (HBS / 4)
                            + (size_t)blockIdx.x * (ELEMS / 4);
  const int iters = ELEMS / (NT * 4);  // 64
  for (int it = 0; it < iters; ++it) {
    const int vi = it * NT + tid;
    if (it + 1 < iters) __builtin_prefetch(src + vi + NT, 0, 0);  // global_prefetch_b8
    float4 v = src[vi];
    atomicAdd(&lh[order_map(__float_as_uint(v.x)) >> BIN_SH], 1u);
    atomicAdd(&lh[order_map(__float_as_uint(v.y)) >> BIN_SH], 1u);
    atomicAdd(&lh[order_map(__float_as_uint(v.z)) >> BIN_SH], 1u);
    atomicAdd(&lh[order_map(__float_as_uint(v.w)) >> BIN_SH], 1u);
  }
  __syncthreads();

  u32* gh = ghist + (size_t)batch * NBINS;
  for (int i = tid; i < NBINS; i += NT) {
    u32 c = lh[i];
    if (c) atomicAdd(&gh[i], c);
  }
}

// ---------------------------------------------------------------------------
// Find threshold bin per batch: first bin (scanning from the top) where the
// cumulative count reaches TOPK.  grid = (B), block = NT.
// Histogram is destined for LDS -> use async global->LDS copy (b128).
// ---------------------------------------------------------------------------
__global__ __launch_bounds__(NT)
void thresh_kernel(const u32* __restrict__ ghist, u32* __restrict__ bstar) {
  __shared__ u32 lh[NBINS];
  __shared__ u32 csum[NT];
  const int tid = threadIdx.x;
  const int batch = blockIdx.x;
  const u32* gh = ghist + (size_t)batch * NBINS;

#if HAS_ASYNC_LDS
  for (int vi = tid; vi < NBINS / 4; vi += NT) {
    __builtin_amdgcn_global_load_async_to_lds_b128(
        (gv4i*)(gh + 4 * vi), (lv4i*)&lh[4 * vi], 0, 0);
  }
  __builtin_amdgcn_s_wait_asynccnt(0);
#else
  for (int i = tid; i < NBINS; i += NT) lh[i] = gh[i];
#endif
  __syncthreads();

  // chunk tid covers bins [hi-31 .. hi] in the descending scan order
  {
    u32 s = 0;
    const int hi = NBINS - 1 - 32 * tid;
    for (int k = 0; k < 32; ++k) s += lh[hi - k];
    csum[tid] = s;
  }
  __syncthreads();

  if (tid == 0) {
    u32 run = 0, bs = 0;
    for (int q = 0; q < NT; ++q) {
      const u32 cs = csum[q];
      if (run + cs >= (u32)TOPK) {
        const int h2 = NBINS - 1 - 32 * q;
        for (int k = 0; k < 32; ++k) {
          run += lh[h2 - k];
          if (run >= (u32)TOPK) { bs = (u32)(h2 - k); break; }
        }
        break;
      }
      run += cs;
    }
    bstar[batch] = bs;
  }
}

// ---------------------------------------------------------------------------
// Pass 2: collect every element whose bin >= B* as a packed 64-bit key:
//   key = (order_map(bits) << 32) | ~index
// ~index gives jax.lax.top_k's smaller-index-first tie-break under a
// descending key sort.  grid = (BPB, B), block = NT
// ---------------------------------------------------------------------------
__global__ __launch_bounds__(NT)
void collect_kernel(const float4* __restrict__ heat4, const u32* __restrict__ bstar,
                    u32* __restrict__ cnt, u64* __restrict__ cand) {
  const int tid = threadIdx.x;
  const int batch = blockIdx.y;
  const u32 thresh = bstar[batch] << BIN_SH;   // m >= thresh  <=>  bin >= B*
  u32* bc = &cnt[batch];
  u64* cb = cand + (size_t)batch * CAP;

  const float4* src = heat4 + (size_t)batch * (HBS / 4)
                            + (size_t)blockIdx.x * (ELEMS / 4);
  const u32 base_idx = (u32)blockIdx.x * (u32)ELEMS;
  const int iters = ELEMS / (NT * 4);
  for (int it = 0; it < iters; ++it) {
    const int vi = it * NT + tid;
    if (it + 1 < iters) __builtin_prefetch(src + vi + NT, 0, 0);
    float4 v = src[vi];
    const u32 i0 = base_idx + (u32)vi * 4u;
    u32 m;
    m = order_map(__float_as_uint(v.x));
    if (m >= thresh) { u32 p = atomicAdd(bc, 1u); if (p < CAP) cb[p] = ((u64)m << 32) | (u32)~(i0 + 0u); }
    m = order_map(__float_as_uint(v.y));
    if (m >= thresh) { u32 p = atomicAdd(bc, 1u); if (p < CAP) cb[p] = ((u64)m << 32) | (u32)~(i0 + 1u); }
    m = order_map(__float_as_uint(v.z));
    if (m >= thresh) { u32 p = atomicAdd(bc, 1u); if (p < CAP) cb[p] = ((u64)m << 32) | (u32)~(i0 + 2u); }
    m = order_map(__float_as_uint(v.w));
    if (m >= thresh) { u32 p = atomicAdd(bc, 1u); if (p < CAP) cb[p] = ((u64)m << 32) | (u32)~(i0 + 3u); }
  }
}

// ---------------------------------------------------------------------------
// Per-batch: bitonic sort (descending) of up to CAP candidate keys in LDS,
// emit sorted top KEEP.  grid = (B), block = NT, 64 KB LDS.
// Candidates are destined for LDS -> async global->LDS (b64) for live slots;
// padding slots get ds-store zeros (disjoint addresses, ordered by the
// explicit s_wait_asynccnt + barrier).  Canonicalizes atomic-append order.
// ---------------------------------------------------------------------------
__global__ __launch_bounds__(NT)
void sort_kernel(const u64* __restrict__ cand, const u32* __restrict__ cnt,
                 u64* __restrict__ finalk) {
  __shared__ u64 lds[CAP];
  const int tid = threadIdx.x;
  const int batch = blockIdx.x;
  u32 n = cnt[batch];
  if (n > CAP) n = CAP;
  const u64* cb = cand + (size_t)batch * CAP;

  for (int i = tid; i < CAP; i += NT) {
    if (i < (int)n) {
#if HAS_ASYNC_LDS
      __builtin_amdgcn_global_load_async_to_lds_b64(
          (gv2i*)(cb + i), (lv2i*)&lds[i], 0, 0);
#else
      lds[i] = cb[i];
#endif
    } else {
      lds[i] = 0ull;
    }
  }
#if HAS_ASYNC_LDS
  __builtin_amdgcn_s_wait_asynccnt(0);
#endif
  __syncthreads();

  for (u32 k = 2; k <= (u32)CAP; k <<= 1) {
    for (u32 j = k >> 1; j > 0; j >>= 1) {
      for (u32 i = (u32)tid; i < (u32)CAP; i += NT) {
        const u32 l = i ^ j;
        if (l > i) {
          u64 a = lds[i], b2 = lds[l];
          const bool desc = ((i & k) == 0);
          if (desc ? (a < b2) : (a > b2)) { lds[i] = b2; lds[l] = a; }
        }
      }
      __syncthreads();
    }
  }

  u64* fk = finalk + (size_t)batch * KEEP;
  for (int t = tid; t < KEEP; t += NT) fk[t] = lds[t];
}

// ---------------------------------------------------------------------------
// Decode top-100 keys, gather offset channels, mask rows below MIN_CONF,
// write [B, TOPK, 6] = {y, x, height, width, class, conf}. grid=(B), block=128
// ---------------------------------------------------------------------------
__global__ __launch_bounds__(128)
void gather_kernel(const float* __restrict__ heat, const u64* __restrict__ fin,
                   float* __restrict__ out) {
  const int b = blockIdx.x;
  const int t = threadIdx.x;
  if (t >= TOPK) return;
  const u64 key = fin[(size_t)b * KEEP + t];
  const u32 m = (u32)(key >> 32);
  const u32 vb = (m & 0x80000000u) ? (m & 0x7FFFFFFFu) : ~m;  // inverse order_map
  float conf = __uint_as_float(vb);
  const u32 idx = ~(u32)(key & 0xFFFFFFFFu);
  const u32 cls = idx >> 16;        // / HW
  const u32 rem = idx & 0xFFFFu;    // % HW
  const u32 iy = rem >> 8;          // / W
  const u32 ix = rem & 0xFFu;       // % W

  const float* hb = heat + (size_t)b * HBS;
  float y  = (float)iy + hb[(size_t)(C_CLS + 0) * HW + rem];
  float x  = (float)ix + hb[(size_t)(C_CLS + 1) * HW + rem];
  float hh =             hb[(size_t)(C_CLS + 2) * HW + rem];
  float ww =             hb[(size_t)(C_CLS + 3) * HW + rem];
  float cf = (float)cls;
  if (conf < MIN_CONF) { y = 0.f; x = 0.f; hh = 0.f; ww = 0.f; cf = 0.f; conf = 0.f; }

  float* o = out + ((size_t)b * TOPK + t) * 6;
  o[0] = y; o[1] = x; o[2] = hh; o[3] = ww; o[4] = cf; o[5] = conf;
}

// ---------------------------------------------------------------------------
extern "C" void kernel_launch(void* const* d_in, const int* in_sizes, int n_in,
                              void* d_out, int out_size, void* d_ws, size_t ws_size,
                              hipStream_t stream) {
  (void)in_sizes; (void)n_in; (void)out_size; (void)ws_size;
  const float* heat = (const float*)d_in[0];
  float* out = (float*)d_out;

  // Workspace layout (total ~1.52 MB):
  char* ws = (char*)d_ws;
  u32* ghist = (u32*)ws;                                   // 16*8192*4 = 524288 B
  u32* cnt   = (u32*)(ws + 524288);                        // 64 B
  u32* bstar = (u32*)(ws + 524288 + 64);                   // 64 B
  u64* cand  = (u64*)(ws + 524288 + 128);                  // 16*8192*8 = 1 MB
  u64* fin   = (u64*)(ws + 524288 + 128 + 1048576);        // 16*128*8 = 16 KB

  // Zero histogram + per-batch counters every call (graph-capture safe).
  (void)hipMemsetAsync(ws, 0, 524288 + 64, stream);

  hist_kernel   <<<dim3(BPB, B), NT,  0, stream>>>((const float4*)heat, ghist);
  thresh_kernel <<<dim3(B),      NT,  0, stream>>>(ghist, bstar);
  collect_kernel<<<dim3(BPB, B), NT,  0, stream>>>((const float4*)heat, bstar, cnt, cand);
  sort_kernel   <<<dim3(B),      NT,  0, stream>>>(cand, cnt, fin);
  gather_kernel <<<dim3(B),      128, 0, stream>>>(heat, fin, out);
}